// EdgePoint2Wrapper_32744830665316
// MI455X (gfx1250) — compile-verified
//
#include <hip/hip_runtime.h>
#include <stdint.h>

#define HBINS 8192
#define CAND_CAP 65536
#define COMPACT_CAP 8192
#define TOPK 4096
#define IMG_W 1024
#define IMG_H 1024

#define TILE_W 64
#define TILE_H 16
#define LDS_W 72              // global cols [x0-4, x0+67], 16B-aligned chunks
#define LDS_H 20              // global rows [y0-2, y0+17]
#define CHUNKS_PER_ROW (LDS_W / 4)
#define NCHUNK (LDS_H * CHUNKS_PER_ROW)   // 360 x 16B async loads per tile

// ---- CDNA5 async global->LDS path (ASYNCcnt), with sync fallback ----------
#if defined(__has_builtin)
#if __has_builtin(__builtin_amdgcn_global_load_async_to_lds_b128) && \
    __has_builtin(__builtin_amdgcn_s_wait_asynccnt)
#define USE_ASYNC_LDS 1
#endif
#endif

typedef int ep2_v4i __attribute__((ext_vector_type(4)));
typedef __attribute__((address_space(1))) ep2_v4i ep2_gv4i;  // global
typedef __attribute__((address_space(3))) ep2_v4i ep2_lv4i;  // LDS

// ---------------------------------------------------------------------------
__global__ __launch_bounds__(256) void ep2_init_kernel(unsigned* counters,
                                                       unsigned* hist) {
  int i = blockIdx.x * 256 + threadIdx.x;
  if (i < 4) counters[i] = 0u;
  if (i < HBINS) hist[i] = 0u;
}

// One block per 64x16 output tile. Stage (72x20) apron tile into LDS with
// async b128 loads, then 5x5 NMS per pixel. Detected pixels append a 64-bit
// sort key (scoreBits<<32 | ~flatIdx) and bump a value-histogram bin.
__global__ __launch_bounds__(256) void ep2_detect_kernel(
    const float* __restrict__ score, unsigned long long* __restrict__ cand,
    unsigned* __restrict__ counters, unsigned* __restrict__ hist) {
  __shared__ float tile[LDS_H][LDS_W];
  const int x0 = blockIdx.x * TILE_W;
  const int y0 = blockIdx.y * TILE_H;
  const int t = threadIdx.x;

  // Stage tile: NCHUNK 16-byte chunks; skip chunks mapping outside the image
  // (only border-masked pixels can ever read those LDS cells).
  for (int c = t; c < NCHUNK; c += 256) {
    const int row = c / CHUNKS_PER_ROW;
    const int col4 = (c % CHUNKS_PER_ROW) * 4;
    const int gy = y0 - 2 + row;
    const int gx = x0 - 4 + col4;          // multiple of 4 -> 16B aligned
    if (gy >= 0 && gy < IMG_H && gx >= 0 && gx + 3 < IMG_W) {
#ifdef USE_ASYNC_LDS
      __builtin_amdgcn_global_load_async_to_lds_b128(
          (ep2_gv4i*)(uintptr_t)(score + gy * IMG_W + gx),
          (ep2_lv4i*)&tile[row][col4],
          /*offset=*/0, /*cpol=*/0);
#else
      const float4 v = *(const float4*)(score + gy * IMG_W + gx);
      *(float4*)&tile[row][col4] = v;
#endif
    }
  }
#ifdef USE_ASYNC_LDS
  __builtin_amdgcn_s_wait_asynccnt(0);
#endif
  __syncthreads();

  const int tx = t & 63;        // 0..63
  const int ty = t >> 6;        // 0..3, pixel rows ty, ty+4, ty+8, ty+12
  const int lx = tx + 4;        // LDS col of this pixel
#pragma unroll
  for (int rr = 0; rr < 4; ++rr) {
    const int ly = ty + rr * 4;       // 0..15
    const int lyy = ly + 2;           // LDS row of this pixel
    const int gx = x0 + tx;
    const int gy = y0 + ly;
    const float s = tile[lyy][lx];
    float mx = -__builtin_inff();
#pragma unroll
    for (int dy = -2; dy <= 2; ++dy)
#pragma unroll
      for (int dx = -2; dx <= 2; ++dx)
        mx = fmaxf(mx, tile[lyy + dy][lx + dx]);
    const bool det = (s == mx) && (s > 0.0f) && (gx >= 4) &&
                     (gx < IMG_W - 4) && (gy >= 4) && (gy < IMG_H - 4);
    if (det) {
      const unsigned bits = __float_as_uint(s);     // s>0 -> order-preserving
      const unsigned idx = (unsigned)(gy * IMG_W + gx);
      const unsigned long long key =
          ((unsigned long long)bits << 32) | (0xFFFFFFFFu - idx);
      const unsigned p = atomicAdd(&counters[0], 1u);
      if (p < CAND_CAP) cand[p] = key;
      int bin = (int)(s * (float)HBINS);
      bin = bin > (HBINS - 1) ? (HBINS - 1) : (bin < 0 ? 0 : bin);
      atomicAdd(&hist[bin], 1u);
    }
  }
}

// Single block: find largest bin T with suffix-count >= TOPK (else 0); zero compact.
__global__ __launch_bounds__(1024) void ep2_scan_kernel(
    const unsigned* __restrict__ hist, unsigned* counters,
    unsigned long long* compact) {
  const int t = threadIdx.x;
  for (int i = t; i < COMPACT_CAP; i += 1024) compact[i] = 0ull;
  if (t == 0) {
    unsigned acc = 0;
    int thr = 0;
    for (int b = HBINS - 1; b >= 0; --b) {
      acc += hist[b];
      if (acc >= TOPK) { thr = b; break; }
    }
    counters[2] = (unsigned)thr;
    counters[1] = 0u;
  }
}

__global__ __launch_bounds__(256) void ep2_compact_kernel(
    const unsigned long long* __restrict__ cand, unsigned* counters,
    unsigned long long* __restrict__ compact) {
  unsigned n = counters[0];
  if (n > CAND_CAP) n = CAND_CAP;
  const unsigned i = blockIdx.x * 256 + threadIdx.x;
  if (i >= n) return;
  const unsigned long long key = cand[i];
  const float s = __uint_as_float((unsigned)(key >> 32));
  int bin = (int)(s * (float)HBINS);
  bin = bin > (HBINS - 1) ? (HBINS - 1) : (bin < 0 ? 0 : bin);
  if ((unsigned)bin >= counters[2]) {
    const unsigned p = atomicAdd(&counters[1], 1u);
    if (p < COMPACT_CAP) compact[p] = key;
  }
}

// Single block, 64KB LDS: bitonic sort 8192 keys ascending, emit top 4096
// descending. out layout: kpts[4096][2] | scores[4096] | descs[4096][64].
__global__ __launch_bounds__(1024) void ep2_sort_kernel(
    const unsigned long long* __restrict__ compact, float* __restrict__ out,
    unsigned* __restrict__ sel_idx) {
  __shared__ unsigned long long k[COMPACT_CAP];
  const int t = threadIdx.x;
  for (int i = t; i < COMPACT_CAP; i += 1024) k[i] = compact[i];
  __syncthreads();
  for (int kk = 2; kk <= COMPACT_CAP; kk <<= 1) {
    for (int j = kk >> 1; j > 0; j >>= 1) {
      for (int i = t; i < COMPACT_CAP; i += 1024) {
        const int ixj = i ^ j;
        if (ixj > i) {
          const bool up = ((i & kk) == 0);
          const unsigned long long a = k[i];
          const unsigned long long b = k[ixj];
          if ((a > b) == up) { k[i] = b; k[ixj] = a; }
        }
      }
      __syncthreads();
    }
  }
  for (int r = t; r < TOPK; r += 1024) {
    const unsigned long long key = k[COMPACT_CAP - 1 - r];
    const unsigned bits = (unsigned)(key >> 32);
    unsigned idx = 0xFFFFFFFFu - (unsigned)(key & 0xFFFFFFFFull);
    if (idx >= (unsigned)(IMG_W * IMG_H)) idx = 0u;  // padding guard
    out[r * 2 + 0] = (float)(idx & (IMG_W - 1));     // kx
    out[r * 2 + 1] = (float)(idx >> 10);             // ky
    out[2 * TOPK + r] = __uint_as_float(bits);       // score
    sel_idx[r] = idx;
  }
}

// descs[kp][c] = raw_desc[c][idx]; bilinear collapses to an exact gather
// because px == kx, py == ky for this reference's coordinate transform.
__global__ __launch_bounds__(256) void ep2_gather_kernel(
    const float* __restrict__ desc, const unsigned* __restrict__ sel_idx,
    float* __restrict__ out_desc) {
  const int t = blockIdx.x * 256 + threadIdx.x;  // over TOPK*64
  const int kp = t >> 6;
  const int c = t & 63;
  const unsigned idx = sel_idx[kp];
  out_desc[kp * 64 + c] =
      desc[(size_t)c * (size_t)(IMG_W * IMG_H) + (size_t)idx];
}

// ---------------------------------------------------------------------------
extern "C" void kernel_launch(void* const* d_in, const int* in_sizes, int n_in,
                              void* d_out, int out_size, void* d_ws,
                              size_t ws_size, hipStream_t stream) {
  const float* raw_desc = (const float*)d_in[0];    // [64,1024,1024]
  const float* raw_detect = (const float*)d_in[1];  // [1024,1024]
  float* out = (float*)d_out;                       // 274432 floats

  char* ws = (char*)d_ws;
  unsigned* counters = (unsigned*)ws;                                  // 16 B
  unsigned* hist = (unsigned*)(ws + 16);                               // 32 KB
  unsigned long long* cand = (unsigned long long*)(ws + 16 + HBINS * 4);
  unsigned long long* compact = cand + CAND_CAP;                       // 64 KB
  unsigned* sel_idx = (unsigned*)(compact + COMPACT_CAP);              // 16 KB

  ep2_init_kernel<<<(HBINS + 255) / 256, 256, 0, stream>>>(counters, hist);
  ep2_detect_kernel<<<dim3(IMG_W / TILE_W, IMG_H / TILE_H), 256, 0, stream>>>(
      raw_detect, cand, counters, hist);
  ep2_scan_kernel<<<1, 1024, 0, stream>>>(hist, counters, compact);
  ep2_compact_kernel<<<CAND_CAP / 256, 256, 0, stream>>>(cand, counters,
                                                         compact);
  ep2_sort_kernel<<<1, 1024, 0, stream>>>(compact, out, sel_idx);
  ep2_gather_kernel<<<(TOPK * 64) / 256, 256, 0, stream>>>(raw_desc, sel_idx,
                                                           out + 3 * TOPK);
}